// ClusterDiceLoss_30185030156401
// MI455X (gfx1250) — compile-verified
//
#include <hip/hip_runtime.h>
#include <hip/hip_bf16.h>

typedef __attribute__((ext_vector_type(4))) float v4f;
typedef __attribute__((ext_vector_type(4))) int   v4i;
typedef __attribute__((ext_vector_type(2))) float v2f;
typedef __attribute__((ext_vector_type(8))) float v8f;

#define NSEG 65                 // bin 0 = background (dropped), 1..64 = clusters
#define WAVES_PER_BLOCK 8
#define THREADS 256
#define MASK21 0x1FFFFFull
#define CNTRS (3 * NSEG)        // [0..64]=sum_p, [65..129]=sum_t, [130..194]=inter

// ---------------------------------------------------------------- zero scratch
__global__ void ClusterDice_zero_ws(unsigned* __restrict__ ws, int n) {
    int i = blockIdx.x * blockDim.x + threadIdx.x;
    if (i < n) ws[i] = 0u;
}

// ------------------------------------------------------- streaming segment sum
__global__ __launch_bounds__(THREADS) void ClusterDice_accum(
    const float* __restrict__ pred, const float* __restrict__ targ,
    const int* __restrict__ lab, unsigned* __restrict__ ws, int n, int nrep) {
    __shared__ unsigned long long hist[WAVES_PER_BLOCK * NSEG];
    for (int i = threadIdx.x; i < WAVES_PER_BLOCK * NSEG; i += blockDim.x)
        hist[i] = 0ull;
    __syncthreads();

    unsigned long long* whist = &hist[(threadIdx.x >> 5) * NSEG];  // wave32
    const int nv = n >> 2;
    const v4f* p4 = (const v4f*)pred;
    const v4f* t4 = (const v4f*)targ;
    const v4i* l4 = (const v4i*)lab;
    const int gid    = blockIdx.x * blockDim.x + threadIdx.x;
    const int stride = gridDim.x * blockDim.x;

    for (int i = gid; i < nv; i += stride) {
        v4f pv = __builtin_nontemporal_load(&p4[i]);
        v4f tv = __builtin_nontemporal_load(&t4[i]);
        v4i lv = __builtin_nontemporal_load(&l4[i]);
#pragma unroll
        for (int j = 0; j < 4; ++j) {
            unsigned pi = pv[j] > 0.5f ? 1u : 0u;
            unsigned ti = tv[j] > 0.5f ? 1u : 0u;
            unsigned long long pk = (unsigned long long)pi
                                  | ((unsigned long long)ti << 21)
                                  | ((unsigned long long)(pi & ti) << 42);
            if (pk) atomicAdd(&whist[lv[j]], pk);   // ds_add_u64, per-wave bins
        }
    }
    // scalar tail (n not multiple of 4)
    if (blockIdx.x == 0) {
        for (int i = (nv << 2) + threadIdx.x; i < n; i += blockDim.x) {
            unsigned pi = pred[i] > 0.5f ? 1u : 0u;
            unsigned ti = targ[i] > 0.5f ? 1u : 0u;
            unsigned long long pk = (unsigned long long)pi
                                  | ((unsigned long long)ti << 21)
                                  | ((unsigned long long)(pi & ti) << 42);
            if (pk) atomicAdd(&whist[lab[i]], pk);
        }
    }
    __syncthreads();

    // flush per-block totals into one of nrep replicated global counter sets
    unsigned* base = ws + (blockIdx.x % nrep) * CNTRS;
    for (int b = threadIdx.x; b < NSEG; b += blockDim.x) {
        if (b == 0) continue;                        // drop background segment
        unsigned long long s = 0ull;
#pragma unroll
        for (int w = 0; w < WAVES_PER_BLOCK; ++w) s += hist[w * NSEG + b];
        if (s) {
            atomicAdd(&base[b],            (unsigned)( s        & MASK21)); // sum_p
            atomicAdd(&base[NSEG + b],     (unsigned)((s >> 21) & MASK21)); // sum_t
            atomicAdd(&base[2*NSEG + b],   (unsigned)( s >> 42));           // inter
        }
    }
}

// -------------------------------------------- dice + mean via WMMA (one wave)
__global__ void ClusterDice_final(const unsigned* __restrict__ ws,
                                  float* __restrict__ out, int nrep) {
    __shared__ unsigned tot[CNTRS];
    for (int j = threadIdx.x; j < CNTRS; j += blockDim.x) {
        unsigned acc = 0u;
        for (int r = 0; r < nrep; ++r) acc += ws[r * CNTRS + j];
        tot[j] = acc;
    }
    __syncthreads();

    if (threadIdx.x < 32) {                          // wave 0, EXEC all ones
        int lane = threadIdx.x;
        int m  = lane & 15;                          // A-matrix row
        int kb = (lane >> 4) << 1;                   // K=0/1 (lanes 0-15), K=2/3
        float d[2];
#pragma unroll
        for (int j = 0; j < 2; ++j) {
            int c = m * 4 + kb + j + 1;              // cluster 1..64
            unsigned sp = tot[c];
            unsigned st = tot[NSEG + c];
            unsigned it = tot[2 * NSEG + c];
            unsigned u  = sp + st;
            d[j] = (u > 0u) ? (2.0f * (float)it / (float)u) : 1.0f;
        }
        // A(16x4) = dice values, B(4x16) = ones  =>  D[m,n] = row-sum of dice
        v2f a; a[0] = d[0]; a[1] = d[1];
        v2f b; b[0] = 1.0f; b[1] = 1.0f;
        v8f c = {};
        c = __builtin_amdgcn_wmma_f32_16x16x4_f32(
                false, a, false, b, (short)0, c, false, false);
        float s = c[0] + c[1] + c[2] + c[3] + c[4] + c[5] + c[6] + c[7];
        s += __shfl_xor(s, 16, 32);                  // rows 0-7 + rows 8-15
        if (lane == 0) out[0] = 1.0f - s * (1.0f / 64.0f);
    }
}

// ---------------------------------------------------------------------- launch
extern "C" void kernel_launch(void* const* d_in, const int* in_sizes, int n_in,
                              void* d_out, int out_size, void* d_ws, size_t ws_size,
                              hipStream_t stream) {
    const float* pred = (const float*)d_in[0];
    const float* targ = (const float*)d_in[1];
    const int*   lab  = (const int*)d_in[2];
    unsigned* ws  = (unsigned*)d_ws;
    float*    out = (float*)d_out;
    const int n = in_sizes[0];

    // replicated global counter sets, sized to available scratch (deterministic)
    int nrep = (int)(ws_size / (CNTRS * sizeof(unsigned)));
    if (nrep > 64) nrep = 64;
    if (nrep < 1)  nrep = 1;

    int zn = nrep * CNTRS;
    ClusterDice_zero_ws<<<(zn + 255) / 256, 256, 0, stream>>>(ws, zn);

    int nv = n >> 2;
    int blocks = (nv + THREADS - 1) / THREADS;
    if (blocks > 2048) blocks = 2048;
    if (blocks < 1)    blocks = 1;
    ClusterDice_accum<<<blocks, THREADS, 0, stream>>>(pred, targ, lab, ws, n, nrep);

    ClusterDice_final<<<1, 256, 0, stream>>>(ws, out, nrep);
}